// E3nnMLPBlockS2Grid_63024350102052
// MI455X (gfx1250) — compile-verified
//
#include <hip/hip_runtime.h>
#include <math.h>

typedef __attribute__((ext_vector_type(2))) float v2f;
typedef __attribute__((ext_vector_type(8))) float v8f;

#define MULT 128
#define BATCH 4096
#define GRES 100
#define NP 10000
#define NI 36
#define FEAT 4608
#define INV1 0.08838834764831845f   /* 1/sqrt(128) */
#define INV2 0.0078125f             /* 1/128 */
#define EPSBN 1e-5f
#define PI_D 3.14159265358979323846
#define LOG2E 1.4426950408889634f

__device__ __forceinline__ float fast_sigmoid(float x) {
  return __builtin_amdgcn_rcpf(1.0f + __builtin_amdgcn_exp2f(-LOG2E * x));
}

/* ---------------- K1: Legendre tables + quadrature Vandermonde ---------------- */
__global__ void k_legendre(float* __restrict__ PL, double* __restrict__ AD) {
  int j = threadIdx.x;
  if (j < GRES) {
    double cbd = cos(((double)j + 0.5) * PI_D / GRES);
    float cb = (float)cbd;
    float s = sqrtf(fmaxf(1.0f - cb * cb, 0.0f));
    float P[6][6];
    P[0][0] = 1.0f;
    for (int m = 1; m <= 5; ++m) P[m][m] = -(2.0f * m - 1.0f) * s * P[m - 1][m - 1];
    for (int m = 0; m <= 4; ++m) P[m + 1][m] = (2.0f * m + 1.0f) * cb * P[m][m];
    for (int m = 0; m <= 5; ++m)
      for (int l = m + 2; l <= 5; ++l)
        P[l][m] = ((2.0f * l - 1.0f) * cb * P[l - 1][m] - (l + m - 1.0f) * P[l - 2][m]) / (float)(l - m);
    for (int l = 0; l <= 5; ++l)
      for (int m = 0; m <= l; ++m)
        PL[(l * (l + 1) / 2 + m) * GRES + j] = P[l][m];
    double L0 = 1.0, L1 = cbd;
    AD[0 * 101 + j] = 1.0;
    AD[1 * 101 + j] = cbd;
    for (int n = 2; n < GRES; ++n) {
      double Ln = ((2.0 * n - 1.0) * cbd * L1 - (n - 1.0) * L0) / (double)n;
      AD[n * 101 + j] = Ln;
      L0 = L1; L1 = Ln;
    }
  }
  if (threadIdx.x == 0)
    for (int n = 0; n < GRES; ++n) AD[n * 101 + 100] = (n == 0) ? 2.0 : 0.0;
}

/* ---------------- K2: 100x100 Gauss-Jordan (double) for quadrature weights ---- */
__global__ void k_gauss(double* __restrict__ AD, double* __restrict__ QW) {
  __shared__ int piv;
  int tid = threadIdx.x;
  for (int k = 0; k < GRES; ++k) {
    if (tid == 0) {
      int pm = k; double best = fabs(AD[k * 101 + k]);
      for (int r = k + 1; r < GRES; ++r) {
        double v = fabs(AD[r * 101 + k]);
        if (v > best) { best = v; pm = r; }
      }
      piv = pm;
    }
    __syncthreads();
    int pm = piv;
    if (pm != k)
      for (int c = tid; c <= GRES; c += blockDim.x) {
        double t = AD[k * 101 + c]; AD[k * 101 + c] = AD[pm * 101 + c]; AD[pm * 101 + c] = t;
      }
    __syncthreads();
    double pv = AD[k * 101 + k];
    for (int r = tid; r < GRES; r += blockDim.x) {
      if (r == k) continue;
      double f = AD[r * 101 + k] / pv;
      for (int c = k; c <= GRES; ++c) AD[r * 101 + c] -= f * AD[k * 101 + c];
    }
    __syncthreads();
  }
  if (tid < GRES) QW[tid] = AD[tid * 101 + 100] / AD[tid * 101 + tid];
}

/* ------- K3: build YTO^T [10000][40] and YFROM [36][10000] (b64-friendly) ----- */
__global__ void k_buildY(const float* __restrict__ PL, const double* __restrict__ QW,
                         float* __restrict__ YTOT, float* __restrict__ YF) {
  int t = blockIdx.x * blockDim.x + threadIdx.x;
  if (t >= NP * NI) return;
  int p = t / NI, i = t % NI;
  int l = 0; while ((l + 1) * (l + 1) <= i) ++l;
  int m = i - l * l - l;
  int am = m < 0 ? -m : m;
  int beta = p / GRES, alph = p % GRES;
  double ratio = 1.0;
  for (int q = l - am + 1; q <= l + am; ++q) ratio *= (double)q;
  double Nn = sqrt((2.0 * l + 1.0) / (4.0 * PI_D) / ratio);
  double c = (m == 0) ? 1.0 : sqrt(2.0);
  double av = (double)alph * 2.0 * PI_D / GRES;
  double ang = (m == 0) ? 1.0 : ((m > 0) ? cos(m * av) : sin(am * av));
  double Y = c * Nn * (double)PL[(l * (l + 1) / 2 + am) * GRES + beta] * ang;
  double n_to = sqrt(4.0 * PI_D) * sqrt(2.0 * l + 1.0) / sqrt(6.0);
  YTOT[p * 40 + i] = (float)(Y * n_to);
  YF[i * NP + p] = (float)(Y / n_to * QW[beta] * (2.0 * PI_D / GRES));
}

/* ------- K4: per-l Gram G_l = X_l X_l^T via f32 WMMA (batched loads) ---------- */
template <int LVAL>
__global__ void k_gram_t(const float* __restrict__ x, float* __restrict__ GR) {
  constexpr int d = 2 * LVAL + 1;
  constexpr int off = MULT * LVAL * LVAL;
  constexpr int K16 = BATCH * d / 16;
  constexpr int q = 16 / d, r = 16 - q * d;
  int u0 = (blockIdx.x >> 3) * 16;
  int v0 = (blockIdx.x & 7) * 16;
  int lane = threadIdx.x & 31, wave = threadIdx.x >> 5;
  int row = lane & 15, hi = lane >> 4;
  int kA = wave * 4 + 2 * hi;
  int bA = kA / d, mA = kA - bA * d;
  int bB = (kA + 1) / d, mB = (kA + 1) - bB * d;
  int aA = bA * FEAT + mA;
  int aB = bB * FEAT + mB;
  const float* pu = x + off + (u0 + row) * d;
  const float* pv = x + off + (v0 + row) * d;
  v8f acc = 0;
  for (int it = 0; it < K16 / 4; ++it) {
    v2f af[4], bf[4];
#pragma unroll
    for (int s = 0; s < 4; ++s) {
      af[s].x = pu[aA]; af[s].y = pu[aB];
      bf[s].x = pv[aA]; bf[s].y = pv[aB];
      mA += r; int wA = (mA >= d) ? 1 : 0;
      aA += q * FEAT + r + (wA ? (FEAT - d) : 0);
      mA -= wA ? d : 0;
      mB += r; int wB = (mB >= d) ? 1 : 0;
      aB += q * FEAT + r + (wB ? (FEAT - d) : 0);
      mB -= wB ? d : 0;
    }
#pragma unroll
    for (int s = 0; s < 4; ++s)
      acc = __builtin_amdgcn_wmma_f32_16x16x4_f32(false, af[s], false, bf[s], (short)0, acc, false, false);
  }
  __shared__ float red[4 * 256];
#pragma unroll
  for (int j = 0; j < 8; ++j) red[wave * 256 + j * 32 + lane] = acc[j];
  __syncthreads();
  for (int idx = threadIdx.x; idx < 256; idx += blockDim.x) {
    float sum = red[idx] + red[256 + idx] + red[512 + idx] + red[768 + idx];
    int j = idx >> 5, ln = idx & 31;
    int u = u0 + j + 8 * (ln >> 4), v = v0 + (ln & 15);
    GR[LVAL * 16384 + u * 128 + v] = sum;
  }
}

/* ---------------- K4b: mean of scalar channels ------------------------------- */
__global__ void k_mux(const float* __restrict__ x, float* __restrict__ MUX) {
  int u = threadIdx.x;
  float s = 0.0f;
  for (int b = 0; b < BATCH; ++b) s += x[(size_t)b * FEAT + u];
  MUX[u] = s / (float)BATCH;
}

/* ---------------- K5a: BN scale s[l][v] from Gram ---------------------------- */
__global__ void k_scale(const float* __restrict__ W1, const float* __restrict__ bn_w,
                        const float* __restrict__ GR, const float* __restrict__ MUX,
                        float* __restrict__ MUY, float* __restrict__ SV) {
  int l = blockIdx.x, v = threadIdx.x;
  const float* Gl = GR + l * 16384;
  const float* W  = W1 + l * 16384;
  float acc = 0.0f;
  for (int u = 0; u < 128; ++u) {
    float rs = 0.0f;
    for (int u2 = 0; u2 < 128; ++u2) rs += Gl[u * 128 + u2] * W[u2 * 128 + v];
    acc += W[u * 128 + v] * rs;
  }
  float fn;
  if (l == 0) {
    float muy = 0.0f;
    for (int u = 0; u < 128; ++u) muy += MUX[u] * W[u * 128 + v];
    muy *= INV1;
    MUY[v] = muy;
    fn = acc * INV2 / (float)BATCH - muy * muy;
  } else {
    fn = acc * INV2 / ((float)BATCH * (float)(2 * l + 1));
  }
  SV[l * 128 + v] = rsqrtf(fn + EPSBN) * bn_w[l * 128 + v];
}

/* ---------------- K5b: folded effective weights ------------------------------ */
__global__ void k_weff(const float* __restrict__ W1, const float* __restrict__ W2,
                       const float* __restrict__ bn_b, const float* __restrict__ MUY,
                       const float* __restrict__ SV, float* __restrict__ WE,
                       float* __restrict__ ZC) {
  int l = blockIdx.x, u = threadIdx.x;
  const float* W = W1 + l * 16384;
  float acc = 0.0f;
  for (int v = 0; v < 128; ++v) acc += W[u * 128 + v] * SV[l * 128 + v] * W2[l * 128 + v];
  WE[l * 128 + u] = acc * INV2;
  if (l == 0 && u == 0) {
    float zc = 0.0f;
    for (int v = 0; v < 128; ++v) zc += (bn_b[v] - MUY[v] * SV[v]) * W2[v];
    ZC[0] = zc * INV1;
  }
}

/* ---------------- K6: z[b,lm] = sum_u x * w_eff (LDS-staged) ----------------- */
__global__ void k_z(const float* __restrict__ x, const float* __restrict__ WE,
                    const float* __restrict__ ZC, float* __restrict__ Z) {
  __shared__ __align__(16) float lx[FEAT];
  int b = blockIdx.x;
  const float4* xr = (const float4*)(x + (size_t)b * FEAT);
  for (int e = threadIdx.x; e < FEAT / 4; e += blockDim.x) ((float4*)lx)[e] = xr[e];
  __syncthreads();
  int lane = threadIdx.x & 31, wave = threadIdx.x >> 5;
  for (int lm = wave; lm < NI; lm += 4) {
    int l = 0; while ((l + 1) * (l + 1) <= lm) ++l;
    int m = lm - l * l;
    int d = 2 * l + 1, off = 128 * l * l;
    float s = 0.0f;
    for (int uu = lane; uu < 128; uu += 32) s += lx[off + uu * d + m] * WE[l * 128 + uu];
    for (int sh = 16; sh >= 1; sh >>= 1) s += __shfl_xor(s, sh, 32);
    if (lane == 0) Z[b * NI + lm] = s + (lm == 0 ? ZC[0] : 0.0f);
  }
}

/* ------ K7: fused S2 activation, 2 row-tiles/wave, batched b64 frag loads ---- */
__global__ void k_s2act(const float* __restrict__ Z, const float* __restrict__ YTOT,
                        const float* __restrict__ YF, float* __restrict__ Z2) {
  int lane = threadIdx.x & 31, wave = threadIdx.x >> 5;
  int row = lane & 15, hi = lane >> 4;
  int b0 = (blockIdx.x * 4 + wave) * 32;
  __shared__ __align__(16) float sig[4][2][16][18];
  const v2f* ZV  = (const v2f*)Z;
  const v2f* YTV = (const v2f*)YTOT;
  const v2f* YFV = (const v2f*)YF;
  v2f a1[2][9];
#pragma unroll
  for (int t = 0; t < 2; ++t)
#pragma unroll
    for (int kk = 0; kk < 9; ++kk)
      a1[t][kk] = ZV[(b0 + t * 16 + row) * 18 + 2 * kk + hi];
  v8f zacc[2][3];
#pragma unroll
  for (int t = 0; t < 2; ++t)
#pragma unroll
    for (int nt = 0; nt < 3; ++nt) zacc[t][nt] = 0;
  for (int p0 = 0; p0 < NP; p0 += 16) {
    /* batch all 9 B1 fragments, then 18 WMMAs */
    v2f bb[9];
#pragma unroll
    for (int kk = 0; kk < 9; ++kk) bb[kk] = YTV[(p0 + row) * 20 + 2 * kk + hi];
    v8f g0 = 0, g1 = 0;
#pragma unroll
    for (int kk = 0; kk < 9; ++kk) {
      g0 = __builtin_amdgcn_wmma_f32_16x16x4_f32(false, a1[0][kk], false, bb[kk], (short)0, g0, false, false);
      g1 = __builtin_amdgcn_wmma_f32_16x16x4_f32(false, a1[1][kk], false, bb[kk], (short)0, g1, false, false);
    }
#pragma unroll
    for (int j = 0; j < 8; ++j) {
      g0[j] = fast_sigmoid(g0[j]);
      g1[j] = fast_sigmoid(g1[j]);
    }
#pragma unroll
    for (int j = 0; j < 8; ++j) {
      sig[wave][0][j + 8 * hi][row] = g0[j];
      sig[wave][1][j + 8 * hi][row] = g1[j];
    }
    __syncthreads();
    /* batch all GEMM2 fragments (12 global b64 + 8 LDS b64), then 24 WMMAs */
    v2f b2[4][3], a20[4], a21[4];
#pragma unroll
    for (int kp = 0; kp < 4; ++kp) {
      a20[kp] = ((const v2f*)&sig[wave][0][row][0])[2 * kp + hi];
      a21[kp] = ((const v2f*)&sig[wave][1][row][0])[2 * kp + hi];
#pragma unroll
      for (int nt = 0; nt < 3; ++nt)
        b2[kp][nt] = YFV[(nt * 16 + row) * (NP / 2) + p0 / 2 + 2 * kp + hi];
    }
#pragma unroll
    for (int kp = 0; kp < 4; ++kp)
#pragma unroll
      for (int nt = 0; nt < 3; ++nt) {
        zacc[0][nt] = __builtin_amdgcn_wmma_f32_16x16x4_f32(false, a20[kp], false, b2[kp][nt], (short)0, zacc[0][nt], false, false);
        zacc[1][nt] = __builtin_amdgcn_wmma_f32_16x16x4_f32(false, a21[kp], false, b2[kp][nt], (short)0, zacc[1][nt], false, false);
      }
    __syncthreads();
  }
#pragma unroll
  for (int t = 0; t < 2; ++t)
#pragma unroll
    for (int j = 0; j < 8; ++j) {
      int b = b0 + t * 16 + j + 8 * hi;
      Z2[b * NI + row] = zacc[t][0][j];
      Z2[b * NI + 16 + row] = zacc[t][1][j];
      if (32 + row < NI) Z2[b * NI + 32 + row] = zacc[t][2][j];
    }
}

/* ---------------- K8: expand with W3 ----------------------------------------- */
__global__ void k_out(const float* __restrict__ Z2, const float* __restrict__ W3,
                      float* __restrict__ out) {
  int b = blockIdx.x;
  for (int c = threadIdx.x; c < FEAT; c += blockDim.x) {
    int l = 0; while (128 * (l + 1) * (l + 1) <= c) ++l;
    int r = c - 128 * l * l;
    int d = 2 * l + 1;
    int v = r / d, m = r - v * d;
    out[(size_t)b * FEAT + c] = Z2[b * NI + l * l + m] * W3[l * 128 + v];
  }
}

extern "C" void kernel_launch(void* const* d_in, const int* in_sizes, int n_in,
                              void* d_out, int out_size, void* d_ws, size_t ws_size,
                              hipStream_t stream) {
  const float* x    = (const float*)d_in[0];
  const float* W1   = (const float*)d_in[1];
  const float* W2   = (const float*)d_in[2];
  const float* W3   = (const float*)d_in[3];
  const float* bn_w = (const float*)d_in[4];
  const float* bn_b = (const float*)d_in[5];
  float* out = (float*)d_out;

  float* f = (float*)d_ws;
  float* YTOT = f; f += 400000;            /* [10000][40]          */
  float* YF   = f; f += 480000;            /* [36][10000] (+pad)   */
  float* GR   = f; f += 98304;
  float* PL   = f; f += 2100;
  float* MUX  = f; f += 128;
  float* MUY  = f; f += 128;
  float* SV   = f; f += 768;
  float* WE   = f; f += 768;
  float* ZC   = f; f += 8;
  float* Z    = f; f += 147456;
  float* Z2   = f; f += 147456;
  uintptr_t pa = ((uintptr_t)f + 7) & ~(uintptr_t)7;
  double* AD = (double*)pa;
  double* QW = AD + 100 * 101;

  k_legendre<<<1, 128, 0, stream>>>(PL, AD);
  k_gauss<<<1, 128, 0, stream>>>(AD, QW);
  k_buildY<<<(NP * NI + 255) / 256, 256, 0, stream>>>(PL, QW, YTOT, YF);
  k_gram_t<0><<<64, 128, 0, stream>>>(x, GR);
  k_gram_t<1><<<64, 128, 0, stream>>>(x, GR);
  k_gram_t<2><<<64, 128, 0, stream>>>(x, GR);
  k_gram_t<3><<<64, 128, 0, stream>>>(x, GR);
  k_gram_t<4><<<64, 128, 0, stream>>>(x, GR);
  k_gram_t<5><<<64, 128, 0, stream>>>(x, GR);
  k_mux<<<1, 128, 0, stream>>>(x, MUX);
  k_scale<<<6, 128, 0, stream>>>(W1, bn_w, GR, MUX, MUY, SV);
  k_weff<<<6, 128, 0, stream>>>(W1, W2, bn_b, MUY, SV, WE, ZC);
  k_z<<<BATCH, 128, 0, stream>>>(x, WE, ZC, Z);
  k_s2act<<<32, 128, 0, stream>>>(Z, YTOT, YF, Z2);
  k_out<<<BATCH, 256, 0, stream>>>(Z2, W3, out);
}